// reg_encoder_26680336843515
// MI455X (gfx1250) — compile-verified
//
#include <hip/hip_runtime.h>

typedef float v2f __attribute__((ext_vector_type(2)));
typedef float v4f __attribute__((ext_vector_type(4)));
typedef float v8f __attribute__((ext_vector_type(8)));
typedef unsigned int u32x4 __attribute__((ext_vector_type(4)));
typedef int i32x4 __attribute__((ext_vector_type(4)));
typedef int i32x8 __attribute__((ext_vector_type(8)));

#define N_NODES 50000
#define N_EDGES 800000
#define D_IN    256
#define D_HID   512
#define ROW_TILES (N_NODES / 16)                                   // 3125
#define COL_TILES (D_HID / 16)                                     // 32
#define WPB       8                                                // waves per block
#define ROW_GROUPS ((ROW_TILES + WPB - 1) / WPB)                   // 391

// ---------------------------------------------------------------------------
// Hardware f32 atomic add (global_atomic_add_f32), not a CAS loop.
// ---------------------------------------------------------------------------
__device__ __forceinline__ void atomic_add_f32(float* p, float v) {
#if defined(__AMDGCN__)
    unsafeAtomicAdd(p, v);
#else
    atomicAdd(p, v);
#endif
}

// ---------------------------------------------------------------------------
// Kernel 1: zero scratch (agg + cnt). Re-run every call for determinism.
// ---------------------------------------------------------------------------
__global__ void sage_zero_ws(float* __restrict__ ws, int n) {
    int i = blockIdx.x * blockDim.x + threadIdx.x;
    if (i < n) ws[i] = 0.0f;
}

// ---------------------------------------------------------------------------
// Kernel 2: scatter-sum. One wave32 per edge, 8 features per lane.
// x (51 MB) and agg (51 MB) are both L2-resident (192 MB L2), so the 205M
// float atomics are serviced by L2 atomic units, not HBM.
// ---------------------------------------------------------------------------
__global__ void __launch_bounds__(256)
sage_scatter_accum(const float* __restrict__ x,
                   const int*   __restrict__ edge,    // [2, N_EDGES]
                   float* __restrict__ agg,
                   float* __restrict__ cnt) {
    int wave = (blockIdx.x * blockDim.x + threadIdx.x) >> 5;
    int lane = threadIdx.x & 31;
    if (wave >= N_EDGES) return;
    int src = edge[wave];
    int dst = edge[N_EDGES + wave];
    const float* __restrict__ xs = x   + (size_t)src * D_IN;
    float*       __restrict__ ad = agg + (size_t)dst * D_IN;
#pragma unroll
    for (int j = 0; j < 8; ++j) {
        int c = lane + j * 32;
        atomic_add_f32(&ad[c], xs[c]);
    }
    if (lane == 0) atomic_add_f32(&cnt[dst], 1.0f);
}

// ---------------------------------------------------------------------------
// Kernel 3: mean = agg / max(cnt, 1)   (in place)
// ---------------------------------------------------------------------------
__global__ void sage_finalize_mean(float* __restrict__ agg,
                                   const float* __restrict__ cnt) {
    int i = blockIdx.x * blockDim.x + threadIdx.x;
    if (i >= N_NODES * D_IN) return;
    int row = i >> 8;                          // D_IN == 256
    float c = cnt[row];
    agg[i] = agg[i] * (1.0f / fmaxf(c, 1.0f));
}

// ---------------------------------------------------------------------------
// Kernel 4: out = mean @ W_l + b + x @ W_r via V_WMMA_F32_16X16X4_F32.
//
// Block = 8 waves covering 8 row-tiles x 1 col-tile. The 256x16 slices of
// W_l and W_r for this col-tile (16 KB each) are staged once per block into
// LDS — via the Tensor Data Mover (2D D#, tile 16x256, data_size=4B) when
// the toolchain exposes the builtin, else a cooperative vectorized copy.
// Inner loop per WMMA: one global b64 (A fragment, consecutive K pair) and
// two conflict-free ds_load_b32 (B rows k, k+1 at this lane's column).
//
// fp32 WMMA layouts (ISA 7.12.2):
//   A (16x4): lanes 0-15 rows 0..15 K={0,1}; lanes 16-31 same rows K={2,3}
//   B (4x16): lanes 0-15 cols 0..15 K={0,1}; lanes 16-31 same cols K={2,3}
//   C/D (16x16): VGPR r -> M=r (lanes 0-15), M=r+8 (lanes 16-31)
// ---------------------------------------------------------------------------
__global__ void __launch_bounds__(256)
sage_gemm_wmma(const float* __restrict__ mean,
               const float* __restrict__ x,
               const float* __restrict__ Wl,
               const float* __restrict__ Wr,
               const float* __restrict__ bias,
               float* __restrict__ out) {
    __shared__ float lds_wl[D_IN * 16];     // [k][16] row-major, 16 KB
    __shared__ float lds_wr[D_IN * 16];     // [k][16] row-major, 16 KB

    const int lane    = threadIdx.x & 31;
    const int wv      = threadIdx.x >> 5;
    const int colT    = blockIdx.x;                      // 0..31
    const int rowT    = blockIdx.y * WPB + wv;           // may overrun (tail)
    const int colbase = colT * 16;

#if __has_builtin(__builtin_amdgcn_tensor_load_to_lds) && defined(__AMDGCN__)
    // ---- Tensor Data Mover staging (wave 0 issues both descriptors) ----
    if (wv == 0) {
        unsigned long long gal  = (unsigned long long)(uintptr_t)(Wl + colbase);
        unsigned long long gar  = (unsigned long long)(uintptr_t)(Wr + colbase);
        unsigned int       ldsl = (unsigned int)(uintptr_t)(lds_wl);
        unsigned int       ldsr = (unsigned int)(uintptr_t)(lds_wr);

        // D# group1 (ISA 8.4): data_size=4B(code 2); tensor_dim0=512;
        // tensor_dim1=256; tile_dim0=16; tile_dim1=256; dim0_stride=512.
        i32x8 g1 = { (int)(2u   << 16),     // data_size
                     (int)(512u << 16),     // tensor_dim0[15:0] @ bit48
                     (int)(256u << 16),     // tensor_dim1[15:0] @ bit80
                     (int)(16u  << 16),     // tile_dim0 @ bit112
                     (int)256,              // tile_dim1 @ bit128
                     (int)512,              // tensor_dim0_stride[31:0] @ bit160
                     0, 0 };
        i32x4 gz4 = { 0, 0, 0, 0 };
        i32x8 gz8 = { 0, 0, 0, 0, 0, 0, 0, 0 };

        // D# group0 (ISA 8.3): count=1 | lds_addr | global_addr | type=2
        u32x4 g0l = { 1u, ldsl, (unsigned)(gal & 0xFFFFFFFFu),
                      (unsigned)((gal >> 32) & 0x1FFFFFFu) | (2u << 30) };
        u32x4 g0r = { 1u, ldsr, (unsigned)(gar & 0xFFFFFFFFu),
                      (unsigned)((gar >> 32) & 0x1FFFFFFu) | (2u << 30) };

        __builtin_amdgcn_tensor_load_to_lds(g0l, g1, gz4, gz4, gz8, 0);
        __builtin_amdgcn_tensor_load_to_lds(g0r, g1, gz4, gz4, gz8, 0);
        __builtin_amdgcn_s_wait_tensorcnt(0);
    }
    __syncthreads();
#else
    // ---- Cooperative fallback: thread t copies W row t (64 B, 4x b128) ----
    {
        const int t = threadIdx.x;                       // 0..255 == K row
        const float* __restrict__ gl = Wl + (size_t)t * D_HID + colbase;
        const float* __restrict__ gr = Wr + (size_t)t * D_HID + colbase;
        v4f* dl = (v4f*)&lds_wl[t * 16];
        v4f* dr = (v4f*)&lds_wr[t * 16];
#pragma unroll
        for (int j = 0; j < 4; ++j) {
            dl[j] = *(const v4f*)(gl + 4 * j);
            dr[j] = *(const v4f*)(gr + 4 * j);
        }
    }
    __syncthreads();
#endif

    if (rowT >= ROW_TILES) return;                       // tail waves exit whole

    const int l15   = lane & 15;
    const int khalf = (lane >> 4) << 1;                  // 0 or 2
    const int m     = rowT * 16 + l15;                   // A row for this lane
    const int n     = colbase + l15;                     // B/C col for this lane

    v8f acc = {};
    const float* __restrict__ A    = mean;
    const float* __restrict__ ldsW = lds_wl;
#pragma unroll 1
    for (int pass = 0; pass < 2; ++pass) {
        const float* __restrict__ arow = A + (size_t)m * D_IN;
        for (int k = 0; k < D_IN; k += 4) {
            v2f a = *(const v2f*)(arow + k + khalf);     // one 8B global load
            v2f bm;
            bm.x = ldsW[(k + khalf)     * 16 + l15];     // ds_load_b32
            bm.y = ldsW[(k + khalf + 1) * 16 + l15];     // ds_load_b32
            acc = __builtin_amdgcn_wmma_f32_16x16x4_f32(
                false, a, false, bm, (short)0, acc, false, false);
        }
        A = x; ldsW = lds_wr;
    }

    const float bv    = bias[n];
    const int   mBase = rowT * 16 + ((lane >> 4) << 3);  // +0 or +8
#pragma unroll
    for (int r = 0; r < 8; ++r) {
        out[(size_t)(mBase + r) * D_HID + n] = acc[r] + bv;
    }
}

// ---------------------------------------------------------------------------
// Launch
// ---------------------------------------------------------------------------
extern "C" void kernel_launch(void* const* d_in, const int* in_sizes, int n_in,
                              void* d_out, int out_size, void* d_ws, size_t ws_size,
                              hipStream_t stream) {
    const float* x    = (const float*)d_in[0];
    const int*   edge = (const int*)  d_in[1];           // [2, N_EDGES]
    const float* Wl   = (const float*)d_in[2];
    const float* Wr   = (const float*)d_in[3];
    const float* b    = (const float*)d_in[4];
    float*       out  = (float*)d_out;

    float* agg = (float*)d_ws;                           // N_NODES * D_IN
    float* cnt = agg + (size_t)N_NODES * D_IN;           // N_NODES

    // 1) zero scratch
    {
        int zn = N_NODES * D_IN + N_NODES;
        sage_zero_ws<<<(zn + 255) / 256, 256, 0, stream>>>((float*)d_ws, zn);
    }
    // 2) scatter-sum (one wave per edge)
    {
        long long threads = (long long)N_EDGES * 32;
        int blocks = (int)((threads + 255) / 256);
        sage_scatter_accum<<<blocks, 256, 0, stream>>>(x, edge, agg, cnt);
    }
    // 3) mean
    {
        int n = N_NODES * D_IN;
        sage_finalize_mean<<<(n + 255) / 256, 256, 0, stream>>>(agg, cnt);
    }
    // 4) fused dual-GEMM + bias: LDS-staged W tiles + fp32 WMMA
    {
        dim3 grid(COL_TILES, ROW_GROUPS);                // 32 x 391
        sage_gemm_wmma<<<grid, 256, 0, stream>>>(agg, x, Wl, Wr, b, out);
    }
}